// myDecoder_ver_2_80796924773037
// MI455X (gfx1250) — compile-verified
//
#include <hip/hip_runtime.h>
#include <hip/hip_bf16.h>

// ---------------------------------------------------------------------------
// Attention-LSTM greedy decoder for MI455X (gfx1250, wave32, WMMA f32 path).
// B=32, S=128, E=H=1024, EMB=512, V=16000, T=100.
// ---------------------------------------------------------------------------

typedef float v2f __attribute__((ext_vector_type(2)));
typedef float v8f __attribute__((ext_vector_type(8)));

#define B_   32
#define S_   128
#define E_   1024
#define H_   1024
#define EMB_ 512
#define V_   16000
#define T_   100
#define KLSTM_ (EMB_ + H_)   // 1536
#define KCOMB_ (2 * E_)      // 2048

// ---------------------------------------------------------------------------
// init: x0 = [prev0=0 | h0], xcomb[:, :H] = h0, c = c0, lastEos = 0
// ---------------------------------------------------------------------------
__global__ __launch_bounds__(256)
void init_kernel(const float* __restrict__ h0, const float* __restrict__ c0,
                 float* __restrict__ x0, float* __restrict__ xcomb,
                 float* __restrict__ cbuf, int* __restrict__ lastEos) {
    int b = blockIdx.x, tid = threadIdx.x;
    for (int k = tid; k < EMB_; k += 256) x0[(size_t)b * KLSTM_ + k] = 0.0f;
    for (int k = tid; k < H_; k += 256) {
        float h = h0[(size_t)b * H_ + k];
        x0[(size_t)b * KLSTM_ + EMB_ + k] = h;
        xcomb[(size_t)b * KCOMB_ + k] = h;
        cbuf[(size_t)b * H_ + k] = c0[(size_t)b * H_ + k];
    }
    if (tid == 0) lastEos[b] = 0;
}

// ---------------------------------------------------------------------------
// keys = enc_hiddens @ W_att^T   :  [4096 x 1024] = [4096 x 1024][1024 x 1024]
// One wave per (16-row tile, 32-col pair of tiles). grid = 256*32 = 8192.
// ---------------------------------------------------------------------------
__global__ __launch_bounds__(32)
void keys_kernel(const float* __restrict__ enc,    // [B*S][E]
                 const float* __restrict__ W_att,  // [H][E]
                 float* __restrict__ keys) {       // [B*S][H]
    int wg = blockIdx.x;
    int mt = wg >> 5;             // 0..255
    int n0 = (wg & 31) * 32;      // col base (pair of 16-col tiles)
    int lane = threadIdx.x;
    int m  = lane & 15;
    int kb = (lane >> 4) * 2;
    int nA = n0 + m, nB = n0 + 16 + m;

    v8f acc0 = {}, acc1 = {};
    const float* arow = enc + (size_t)(mt * 16 + m) * E_ + kb;
    const float* brA  = W_att + (size_t)nA * E_ + kb;
    const float* brB  = W_att + (size_t)nB * E_ + kb;
    for (int k0 = 0; k0 < E_; k0 += 4) {
        v2f a  = *(const v2f*)(arow + k0);
        v2f bA = *(const v2f*)(brA + k0);
        v2f bB = *(const v2f*)(brB + k0);
        acc0 = __builtin_amdgcn_wmma_f32_16x16x4_f32(false, a, false, bA, (short)0, acc0, false, false);
        acc1 = __builtin_amdgcn_wmma_f32_16x16x4_f32(false, a, false, bB, (short)0, acc1, false, false);
    }
    int mb = mt * 16 + ((lane < 16) ? 0 : 8);
#pragma unroll
    for (int r = 0; r < 8; ++r) {
        size_t row = (size_t)(mb + r) * H_;
        keys[row + nA] = acc0[r];
        keys[row + nB] = acc1[r];
    }
}

// ---------------------------------------------------------------------------
// LSTM step: gates = [prev|h] @ [W_ih|W_hh]^T + b ; fused i/f/g/o elementwise.
// One wave per (m tile in {0,1}, 16-col slice jt of H). grid = 2*64 = 128.
// Each wave accumulates 4 gate tiles (cols j, j+H, j+2H, j+3H) so the LSTM
// nonlinearity combines entirely in WMMA accumulator registers.
// ---------------------------------------------------------------------------
__global__ __launch_bounds__(32)
void lstm_step_kernel(const float* __restrict__ xin,   // [B][1536] = [prev|h]
                      float* __restrict__ cbuf,        // [B][H]
                      float* __restrict__ xcomb,       // [B][2048] (writes h half)
                      float* __restrict__ xout,        // [B][1536] (writes h half)
                      const float* __restrict__ W_ih,  // [4H][EMB]
                      const float* __restrict__ W_hh,  // [4H][H]
                      const float* __restrict__ bih,
                      const float* __restrict__ bhh) {
    int wg = blockIdx.x;
    int mt = wg & 1;
    int jt = wg >> 1;             // 0..63
    int lane = threadIdx.x;
    int m  = lane & 15;
    int kb = (lane >> 4) * 2;
    int n  = jt * 16 + m;         // column within H

    v8f acc[4] = {{}, {}, {}, {}};
    const float* arow = xin + (size_t)(mt * 16 + m) * KLSTM_ + kb;

    for (int k0 = 0; k0 < EMB_; k0 += 4) {
        v2f a = *(const v2f*)(arow + k0);
#pragma unroll
        for (int g = 0; g < 4; ++g) {
            v2f bf = *(const v2f*)(W_ih + (size_t)(g * H_ + n) * EMB_ + k0 + kb);
            acc[g] = __builtin_amdgcn_wmma_f32_16x16x4_f32(false, a, false, bf, (short)0, acc[g], false, false);
        }
    }
    for (int k0 = 0; k0 < H_; k0 += 4) {
        v2f a = *(const v2f*)(arow + EMB_ + k0);
#pragma unroll
        for (int g = 0; g < 4; ++g) {
            v2f bf = *(const v2f*)(W_hh + (size_t)(g * H_ + n) * H_ + k0 + kb);
            acc[g] = __builtin_amdgcn_wmma_f32_16x16x4_f32(false, a, false, bf, (short)0, acc[g], false, false);
        }
    }

    float bias[4];
#pragma unroll
    for (int g = 0; g < 4; ++g) {
        int col = g * H_ + n;
        bias[g] = bih[col] + bhh[col];
    }

    int mb = mt * 16 + ((lane < 16) ? 0 : 8);
#pragma unroll
    for (int r = 0; r < 8; ++r) {
        int brow = mb + r;
        float iv = acc[0][r] + bias[0];
        float fv = acc[1][r] + bias[1];
        float gv = acc[2][r] + bias[2];
        float ov = acc[3][r] + bias[3];
        float si = 1.0f / (1.0f + __expf(-iv));
        float sf = 1.0f / (1.0f + __expf(-fv));
        float so = 1.0f / (1.0f + __expf(-ov));
        float tg = tanhf(gv);
        size_t cidx = (size_t)brow * H_ + n;
        float cn = sf * cbuf[cidx] + si * tg;
        float hn = so * tanhf(cn);
        cbuf[cidx] = cn;
        xcomb[(size_t)brow * KCOMB_ + n] = hn;
        xout[(size_t)brow * KLSTM_ + EMB_ + n] = hn;
    }
}

// ---------------------------------------------------------------------------
// Attention: scores = keys[b,s,:].h ; masked softmax ; ctx = alpha.enc
// One WG (128 threads) per batch row. Small VALU work, LDS reductions.
// ---------------------------------------------------------------------------
__global__ __launch_bounds__(128)
void attn_kernel(const float* __restrict__ keys,     // [B*S][H]
                 float* __restrict__ xcomb,          // [B][2048] (h in, ctx out)
                 const float* __restrict__ enc,      // [B*S][E]
                 const int* __restrict__ enc_len) {
    __shared__ float hsm[H_];
    __shared__ float sc[S_];
    __shared__ float red[S_];
    int b = blockIdx.x, tid = threadIdx.x;

    for (int k = tid; k < H_; k += 128) hsm[k] = xcomb[(size_t)b * KCOMB_ + k];
    __syncthreads();

    // score for s = tid
    {
        const float* krow = keys + (size_t)(b * S_ + tid) * H_;
        float acc = 0.0f;
        for (int k = 0; k < H_; ++k) acc = fmaf(krow[k], hsm[k], acc);
        sc[tid] = (tid < enc_len[b]) ? acc : -1e9f;
    }
    __syncthreads();

    // max reduce
    red[tid] = sc[tid];
    __syncthreads();
    for (int off = 64; off > 0; off >>= 1) {
        if (tid < off) red[tid] = fmaxf(red[tid], red[tid + off]);
        __syncthreads();
    }
    float mx = red[0];
    __syncthreads();

    // exp + sum reduce
    float ev = __expf(sc[tid] - mx);
    red[tid] = ev;
    __syncthreads();
    for (int off = 64; off > 0; off >>= 1) {
        if (tid < off) red[tid] += red[tid + off];
        __syncthreads();
    }
    float inv = 1.0f / red[0];
    __syncthreads();
    sc[tid] = ev * inv;     // alpha
    __syncthreads();

    // ctx[e] = sum_s alpha[s] * enc[b,s,e]
    for (int e = tid; e < E_; e += 128) {
        float acc = 0.0f;
        const float* col = enc + (size_t)b * S_ * E_ + e;
        for (int s = 0; s < S_; ++s) acc = fmaf(sc[s], col[(size_t)s * E_], acc);
        xcomb[(size_t)b * KCOMB_ + E_ + e] = acc;
    }
}

// ---------------------------------------------------------------------------
// Output projection: logits = tanh(xcomb @ W_comb^T)  → out[b][t][:]
// One wave per (m tile, pair of 16-col tiles). grid = 2*500 = 1000.
// ---------------------------------------------------------------------------
__global__ __launch_bounds__(32)
void dec_gemm_kernel(const float* __restrict__ xcomb,  // [B][2048]
                     const float* __restrict__ W_comb, // [V][2048]
                     float* __restrict__ out, int t) {
    int wg = blockIdx.x;
    int mt = wg & 1;
    int n0 = (wg >> 1) * 32;
    int lane = threadIdx.x;
    int m  = lane & 15;
    int kb = (lane >> 4) * 2;
    int nA = n0 + m, nB = n0 + 16 + m;

    v8f acc0 = {}, acc1 = {};
    const float* arow = xcomb + (size_t)(mt * 16 + m) * KCOMB_ + kb;
    const float* brA  = W_comb + (size_t)nA * KCOMB_ + kb;
    const float* brB  = W_comb + (size_t)nB * KCOMB_ + kb;
    for (int k0 = 0; k0 < KCOMB_; k0 += 4) {
        v2f a  = *(const v2f*)(arow + k0);
        v2f bA = *(const v2f*)(brA + k0);
        v2f bB = *(const v2f*)(brB + k0);
        acc0 = __builtin_amdgcn_wmma_f32_16x16x4_f32(false, a, false, bA, (short)0, acc0, false, false);
        acc1 = __builtin_amdgcn_wmma_f32_16x16x4_f32(false, a, false, bB, (short)0, acc1, false, false);
    }
    int mb = mt * 16 + ((lane < 16) ? 0 : 8);
#pragma unroll
    for (int r = 0; r < 8; ++r) {
        size_t base = ((size_t)(mb + r) * T_ + t) * V_;
        out[base + nA] = tanhf(acc0[r]);
        out[base + nB] = tanhf(acc1[r]);
    }
}

// ---------------------------------------------------------------------------
// Finalize per step: argmax over V, log_softmax in-place, greedy embedding
// feedback into xout[:, :EMB], EOS bookkeeping, tgt_len at the last step.
// One WG (256 threads) per batch row.
// ---------------------------------------------------------------------------
__global__ __launch_bounds__(256)
void finalize_kernel(float* __restrict__ out,          // logP buffer [B][T][V] (+ int tail)
                     const float* __restrict__ emb,    // [V][EMB]
                     float* __restrict__ xout,         // [B][1536]
                     int* __restrict__ lastEos, int t) {
    __shared__ float rv[256];
    __shared__ int   ri[256];
    int b = blockIdx.x, tid = threadIdx.x;
    size_t base = ((size_t)b * T_ + t) * V_;

    // local argmax (first occurrence on ties within a thread's stride)
    float lmax = -3.402823e38f;
    int lidx = 0;
    for (int v = tid; v < V_; v += 256) {
        float x = out[base + v];
        if (x > lmax) { lmax = x; lidx = v; }
    }
    rv[tid] = lmax; ri[tid] = lidx;
    __syncthreads();
    for (int off = 128; off > 0; off >>= 1) {
        if (tid < off) {
            float a = rv[tid], c = rv[tid + off];
            if (c > a || (c == a && ri[tid + off] < ri[tid])) {
                rv[tid] = c; ri[tid] = ri[tid + off];
            }
        }
        __syncthreads();
    }
    float M = rv[0];
    int sym = ri[0];
    __syncthreads();

    // sum of exp(x - M)
    float lsum = 0.0f;
    for (int v = tid; v < V_; v += 256) lsum += __expf(out[base + v] - M);
    rv[tid] = lsum;
    __syncthreads();
    for (int off = 128; off > 0; off >>= 1) {
        if (tid < off) rv[tid] += rv[tid + off];
        __syncthreads();
    }
    float lse = __logf(rv[0]);
    __syncthreads();

    // in-place log_softmax
    for (int v = tid; v < V_; v += 256) out[base + v] = out[base + v] - M - lse;

    // greedy embedding feedback
    for (int e = tid; e < EMB_; e += 256)
        xout[(size_t)b * KLSTM_ + e] = emb[(size_t)sym * EMB_ + e];

    // EOS / tgt_len bookkeeping
    if (tid == 0) {
        if (sym == 2) lastEos[b] = t + 1;
        if (t == T_ - 1) {
            int tl = (sym == 2) ? lastEos[b] : T_;
            ((int*)out)[(size_t)B_ * T_ * V_ + b] = tl;
        }
    }
}

// ---------------------------------------------------------------------------
extern "C" void kernel_launch(void* const* d_in, const int* in_sizes, int n_in,
                              void* d_out, int out_size, void* d_ws, size_t ws_size,
                              hipStream_t stream) {
    const float* h0      = (const float*)d_in[0];
    const float* c0      = (const float*)d_in[1];
    const float* enc     = (const float*)d_in[2];
    const int*   enc_len = (const int*)d_in[3];
    const float* W_att   = (const float*)d_in[4];
    const float* W_ih    = (const float*)d_in[5];
    const float* W_hh    = (const float*)d_in[6];
    const float* b_ih    = (const float*)d_in[7];
    const float* b_hh    = (const float*)d_in[8];
    const float* W_comb  = (const float*)d_in[9];
    const float* emb     = (const float*)d_in[10];
    float* out = (float*)d_out;

    // workspace layout (floats)
    float* ws    = (float*)d_ws;
    float* keys  = ws;                                   // 4096*1024
    float* x0    = keys + (size_t)B_ * S_ * H_;          // 32*1536
    float* x1    = x0 + (size_t)B_ * KLSTM_;             // 32*1536
    float* xcomb = x1 + (size_t)B_ * KLSTM_;             // 32*2048
    float* cbuf  = xcomb + (size_t)B_ * KCOMB_;          // 32*1024
    int*   lastEos = (int*)(cbuf + (size_t)B_ * H_);     // 32

    init_kernel<<<B_, 256, 0, stream>>>(h0, c0, x0, xcomb, cbuf, lastEos);
    keys_kernel<<<(B_ * S_ / 16) * (H_ / 32), 32, 0, stream>>>(enc, W_att, keys);

    for (int t = 0; t < T_; ++t) {
        float* xin  = (t & 1) ? x1 : x0;
        float* xout = (t & 1) ? x0 : x1;
        lstm_step_kernel<<<2 * (H_ / 16), 32, 0, stream>>>(xin, cbuf, xcomb, xout,
                                                           W_ih, W_hh, b_ih, b_hh);
        attn_kernel<<<B_, 128, 0, stream>>>(keys, xcomb, enc, enc_len);
        dec_gemm_kernel<<<2 * (V_ / 32), 32, 0, stream>>>(xcomb, W_comb, out, t);
        finalize_kernel<<<B_, 256, 0, stream>>>(out, emb, xout, lastEos, t);
    }
}